// PointSetAbstraction_15092515078590
// MI455X (gfx1250) — compile-verified
//
#include <hip/hip_runtime.h>
#include <cstdint>
#include <cstddef>

#define NB      8
#define NPTS    8192
#define NFEAT   6
#define NPOINT  1024
#define NSAMP   32
#define RAD2    0.04f          // 0.2^2
#define MROWS   (NB * NPOINT * NSAMP)   // 262144
#define EPSV    1e-5f

typedef __attribute__((ext_vector_type(2))) float v2f;
typedef __attribute__((ext_vector_type(8))) float v8f;

// ---------------------------------------------------------------------------
// CDNA5 async global->LDS copy (ASYNCcnt tracked), 16B per lane per issue.
// LDS offset = low 32 bits of the generic shared-pointer address (ISA 10.2).
// ---------------------------------------------------------------------------
__device__ __forceinline__ void async_copy_b128(void* lds_dst, const void* gsrc, int bytes) {
  uint32_t lds = (uint32_t)(uintptr_t)lds_dst;
  uint64_t g   = (uint64_t)(uintptr_t)gsrc;
  for (int o = (int)threadIdx.x * 16; o < bytes; o += (int)blockDim.x * 16) {
    uint32_t l = lds + (uint32_t)o;
    asm volatile("global_load_async_to_lds_b128 %0, %1, %2"
                 :: "v"(l), "v"((uint32_t)o), "s"(g) : "memory");
  }
  asm volatile("s_wait_asynccnt 0" ::: "memory");
}

// ---------------------------------------------------------------------------
// Kernel 1: farthest point sampling. One block (1024 thr) per batch.
// Each thread owns 8 points in registers; block argmax via wave32 shfl + LDS.
// 2 barriers per iteration: stage-1 slots (swv/swi) and the broadcast cell
// (bfar) never alias, so no trailing barrier is needed.
// ---------------------------------------------------------------------------
__global__ __launch_bounds__(1024) void fps_kernel(const float* __restrict__ xyz,
                                                   int* __restrict__ cidx) {
  const int b = blockIdx.x, t = threadIdx.x;
  const float* xb = xyz + (size_t)b * NPTS * 3;
  float px[8], py[8], pz[8], dd[8];
#pragma unroll
  for (int j = 0; j < 8; ++j) {
    int p = t + 1024 * j;
    px[j] = xb[p * 3 + 0]; py[j] = xb[p * 3 + 1]; pz[j] = xb[p * 3 + 2];
    dd[j] = 1e10f;
  }
  __shared__ float swv[32];
  __shared__ int   swi[32];
  __shared__ int   bfar;
  int far = 0;
  for (int it = 0; it < NPOINT; ++it) {
    if (t == 0) cidx[b * NPOINT + it] = far;
    far = __builtin_amdgcn_readfirstlane(far);             // uniform -> scalar loads
    float cx = xb[far * 3 + 0], cy = xb[far * 3 + 1], cz = xb[far * 3 + 2];
    float bv = -1.f; int bi = 0;
#pragma unroll
    for (int j = 0; j < 8; ++j) {
      float dx = px[j] - cx, dy = py[j] - cy, dz = pz[j] - cz;
      float d = dx * dx + dy * dy + dz * dz;
      dd[j] = fminf(dd[j], d);
      if (dd[j] > bv) { bv = dd[j]; bi = t + 1024 * j; }   // ascending idx: > keeps lowest
    }
    for (int off = 16; off > 0; off >>= 1) {               // wave32 reduction
      float ov = __shfl_down(bv, off);
      int   oi = __shfl_down(bi, off);
      if (ov > bv || (ov == bv && oi < bi)) { bv = ov; bi = oi; }
    }
    if ((t & 31) == 0) { swv[t >> 5] = bv; swi[t >> 5] = bi; }
    __syncthreads();
    if (t < 32) {
      bv = swv[t]; bi = swi[t];
      for (int off = 16; off > 0; off >>= 1) {
        float ov = __shfl_down(bv, off);
        int   oi = __shfl_down(bi, off);
        if (ov > bv || (ov == bv && oi < bi)) { bv = ov; bi = oi; }
      }
      if (t == 0) bfar = bi;
    }
    __syncthreads();
    far = bfar;
  }
}

// ---------------------------------------------------------------------------
// Kernel 2: ball query + gather + concat + zero-pad to 16 cols. Wave/centroid.
// "First 32 indices within radius" == scan in index order with ballot/popc.
// Also writes centroids into d_out.
// ---------------------------------------------------------------------------
__global__ __launch_bounds__(256) void group_kernel(const float* __restrict__ xyz,
                                                    const float* __restrict__ feat,
                                                    const int* __restrict__ cidx,
                                                    float* __restrict__ X0,
                                                    float* __restrict__ out_cent) {
  const int gid  = (int)((blockIdx.x * 256 + threadIdx.x) >> 5);
  const int lane = (int)(threadIdx.x & 31);
  const int b = gid >> 10;
  const float* xb = xyz  + (size_t)b * NPTS * 3;
  const float* fb = feat + (size_t)b * NPTS * NFEAT;
  const int ci = cidx[gid];
  const float cx = xb[ci * 3 + 0], cy = xb[ci * 3 + 1], cz = xb[ci * 3 + 2];
  if (lane == 0) {
    out_cent[(size_t)gid * 3 + 0] = cx;
    out_cent[(size_t)gid * 3 + 1] = cy;
    out_cent[(size_t)gid * 3 + 2] = cz;
  }
  int cnt = 0, first = -1;
  for (int n0 = 0; n0 < NPTS && cnt < NSAMP; n0 += 32) {
    const int n = n0 + lane;
    float dx = xb[n * 3 + 0] - cx, dy = xb[n * 3 + 1] - cy, dz = xb[n * 3 + 2] - cz;
    float d2 = dx * dx + dy * dy + dz * dz;
    bool in = d2 <= RAD2;
    unsigned mask = (unsigned)__ballot(in);
    if (first < 0 && mask) first = n0 + __builtin_ctz(mask);
    if (in) {
      int slot = cnt + __popc(mask & ((1u << lane) - 1u));
      if (slot < NSAMP) {
        float* row = X0 + ((size_t)gid * NSAMP + slot) * 16;
        row[0] = dx; row[1] = dy; row[2] = dz;
#pragma unroll
        for (int j = 0; j < NFEAT; ++j) row[3 + j] = fb[(size_t)n * NFEAT + j];
#pragma unroll
        for (int j = 9; j < 16; ++j) row[j] = 0.f;
      }
    }
    cnt += __popc(mask);
  }
  int filled = cnt < NSAMP ? cnt : NSAMP;
  if (filled < NSAMP) {                       // pad with first in-radius point
    float fdx = xb[first * 3 + 0] - cx, fdy = xb[first * 3 + 1] - cy, fdz = xb[first * 3 + 2] - cz;
    for (int slot = filled + lane; slot < NSAMP; slot += 32) {
      float* row = X0 + ((size_t)gid * NSAMP + slot) * 16;
      row[0] = fdx; row[1] = fdy; row[2] = fdz;
#pragma unroll
      for (int j = 0; j < NFEAT; ++j) row[3 + j] = fb[(size_t)first * NFEAT + j];
#pragma unroll
      for (int j = 9; j < 16; ++j) row[j] = 0.f;
    }
  }
}

// ---------------------------------------------------------------------------
// Kernel 3 (x3): GEMM via V_WMMA_F32_16X16X4_F32 + deterministic BN partials.
// Weights async-copied to LDS, then repacked as interleaved K-pairs (float2)
// so each WMMA step needs one aligned ds_load_b64 for B. The per-lane A strip
// (with fused BN+ReLU of the previous layer) is loaded ONCE into registers and
// reused across all N tiles. Y stores use a base pointer + immediate offsets.
// Block = 8 waves; wave w handles M-tile blockIdx.x*8+w and all N tiles.
// ---------------------------------------------------------------------------
template <int CIN, int CINR, int COUT, bool NORM>
__global__ __launch_bounds__(256) void gemm_bn(const float* __restrict__ X,
                                               const float* __restrict__ W,
                                               const float* __restrict__ Bias,
                                               float* __restrict__ Y,
                                               float* __restrict__ psum,
                                               float* __restrict__ pssq,
                                               const float* __restrict__ scIn,
                                               const float* __restrict__ shIn) {
  __shared__ __align__(16) float raw[CINR * COUT];   // staged W[o][c]
  __shared__ __align__(16) v2f   lwp[(CIN / 2) * COUT]; // lwp[(k/2)*COUT+n] = {W[n][k], W[n][k+1]}
  __shared__ float ssum[8][COUT];
  __shared__ float sssq[8][COUT];
  __shared__ float sscin[CIN];
  __shared__ float sshin[CIN];
  const int tid = threadIdx.x;

  async_copy_b128(raw, W, CINR * COUT * 4);
  if (NORM) {
    for (int c = tid; c < CIN; c += 256) { sscin[c] = scIn[c]; sshin[c] = shIn[c]; }
  }
  __syncthreads();
  for (int i = tid; i < (CIN / 2) * COUT; i += 256) {
    int kp = i / COUT, n = i % COUT;
    int k = 2 * kp;
    v2f w;
    w.x = (k     < CINR) ? raw[n * CINR + k]     : 0.f;
    w.y = (k + 1 < CINR) ? raw[n * CINR + k + 1] : 0.f;
    lwp[i] = w;
  }
  __syncthreads();

  const int wave = tid >> 5, lane = tid & 31;
  const int half = lane >> 4, l16 = lane & 15;
  const size_t m0 = ((size_t)blockIdx.x * 8 + wave) * 16;
  const float* xrow = X + (m0 + l16) * CIN;

  // Per-lane A strip, loaded once (BN+ReLU of previous layer fused here).
  v2f areg[CIN / 4];
#pragma unroll
  for (int s = 0; s < CIN / 4; ++s) {
    const int ka = 4 * s + 2 * half;        // lanes<16: K={4s,4s+1}; lanes>=16: K={4s+2,4s+3}
    float a0 = xrow[ka], a1 = xrow[ka + 1];
    if (NORM) {
      a0 = fmaxf(0.f, a0 * sscin[ka]     + sshin[ka]);
      a1 = fmaxf(0.f, a1 * sscin[ka + 1] + sshin[ka + 1]);
    }
    areg[s].x = a0; areg[s].y = a1;
  }

  float* yp = Y + (m0 + (size_t)(8 * half)) * COUT + l16;  // D rows m = r + 8*half

#pragma unroll
  for (int n0 = 0; n0 < COUT; n0 += 16) {
    const v2f* bcol = lwp + half + n0 + l16;  // (ka>>1) = 2s + half
    v8f acc = {0.f, 0.f, 0.f, 0.f, 0.f, 0.f, 0.f, 0.f};
#pragma unroll
    for (int s = 0; s < CIN / 4; ++s) {
      v2f bvv = bcol[(size_t)(2 * s) * COUT]; // single ds_load_b64
      acc = __builtin_amdgcn_wmma_f32_16x16x4_f32(false, areg[s], false, bvv,
                                                  (short)0, acc, false, false);
    }
    const float bias = Bias[n0 + l16];
    float ps = 0.f, pq = 0.f;
#pragma unroll
    for (int r = 0; r < 8; ++r) {
      float v = acc[r] + bias;
      yp[(size_t)r * COUT + n0] = v;          // base + immediate offset
      ps += v; pq += v * v;
    }
    ps += __shfl_down(ps, 16);                // combine M halves (same channel)
    pq += __shfl_down(pq, 16);
    if (lane < 16) { ssum[wave][n0 + lane] = ps; sssq[wave][n0 + lane] = pq; }
  }
  __syncthreads();
  for (int c = tid; c < COUT; c += 256) {     // fixed-order block partial
    float s = 0.f, q = 0.f;
#pragma unroll
    for (int w = 0; w < 8; ++w) { s += ssum[w][c]; q += sssq[w][c]; }
    psum[(size_t)blockIdx.x * COUT + c] = s;
    pssq[(size_t)blockIdx.x * COUT + c] = q;
  }
}

// ---------------------------------------------------------------------------
// Kernel 4 (x3): fold BN stats into per-channel scale/shift (deterministic).
// ---------------------------------------------------------------------------
__global__ void finalize_stats(const float* __restrict__ psum, const float* __restrict__ pssq,
                               int nblk, float mtot,
                               const float* __restrict__ g, const float* __restrict__ beta,
                               float* __restrict__ scale, float* __restrict__ shift, int cout) {
  const int c = threadIdx.x;
  if (c >= cout) return;
  float s = 0.f, q = 0.f;
  for (int b = 0; b < nblk; ++b) { s += psum[(size_t)b * cout + c]; q += pssq[(size_t)b * cout + c]; }
  float mu  = s / mtot;
  float var = q / mtot - mu * mu;
  float sc  = g[c] * rsqrtf(var + EPSV);
  scale[c] = sc;
  shift[c] = beta[c] - mu * sc;
}

// ---------------------------------------------------------------------------
// Kernel 5: BN+ReLU of layer3 output + max over the 32 samples. Wave/centroid.
// ---------------------------------------------------------------------------
__global__ __launch_bounds__(256) void pool_kernel(const float* __restrict__ Y,
                                                   const float* __restrict__ sc,
                                                   const float* __restrict__ sh,
                                                   float* __restrict__ out) {
  const int gid  = (int)((blockIdx.x * 256 + threadIdx.x) >> 5);
  const int lane = (int)(threadIdx.x & 31);
  const float* y = Y + (size_t)gid * NSAMP * 128;
#pragma unroll
  for (int j = 0; j < 4; ++j) {
    const int c = lane + 32 * j;
    const float s = sc[c], h = sh[c];
    float m = -3.4e38f;
    for (int k = 0; k < NSAMP; ++k)
      m = fmaxf(m, fmaxf(0.f, y[(size_t)k * 128 + c] * s + h));
    out[(size_t)gid * 128 + c] = m;
  }
}

// ---------------------------------------------------------------------------
extern "C" void kernel_launch(void* const* d_in, const int* in_sizes, int n_in,
                              void* d_out, int out_size, void* d_ws, size_t ws_size,
                              hipStream_t stream) {
  (void)in_sizes; (void)n_in; (void)out_size; (void)ws_size;
  const float* xyz  = (const float*)d_in[0];
  const float* feat = (const float*)d_in[1];
  const float* w0 = (const float*)d_in[2];
  const float* b0 = (const float*)d_in[3];
  const float* g0 = (const float*)d_in[4];
  const float* be0 = (const float*)d_in[5];
  const float* w1 = (const float*)d_in[6];
  const float* b1 = (const float*)d_in[7];
  const float* g1 = (const float*)d_in[8];
  const float* be1 = (const float*)d_in[9];
  const float* w2 = (const float*)d_in[10];
  const float* b2 = (const float*)d_in[11];
  const float* g2 = (const float*)d_in[12];
  const float* be2 = (const float*)d_in[13];

  char* ws = (char*)d_ws;
  float* stats = (float*)ws;                          // 3 layers x (scale[128]|shift[128])
  int*   cidx  = (int*)(ws + 8192);                   // 8*1024 ints
  float* parts = (float*)(ws + 65536);                // 2 * 2048 * 128 floats
  float* X0 = (float*)(ws + (size_t)(3u << 20));      // 262144 x 16
  float* Y1 = X0 + (size_t)MROWS * 16;                // 262144 x 64
  float* Y2 = Y1 + (size_t)MROWS * 64;                // 262144 x 64
  float* Y3 = Y2 + (size_t)MROWS * 64;                // 262144 x 128

  float* out_cent = (float*)d_out;
  float* out_pool = out_cent + (size_t)NB * NPOINT * 3;

  const int nblk = 2048;                              // 16384 M-tiles / 8 waves
  const float mtot = (float)MROWS;

  fps_kernel<<<dim3(NB), dim3(1024), 0, stream>>>(xyz, cidx);
  group_kernel<<<dim3((NB * NPOINT) / 8), dim3(256), 0, stream>>>(xyz, feat, cidx, X0, out_cent);

  // layer 0: 16(pad of 9) -> 64
  gemm_bn<16, 9, 64, false><<<dim3(nblk), dim3(256), 0, stream>>>(
      X0, w0, b0, Y1, parts, parts + (size_t)nblk * 64, nullptr, nullptr);
  finalize_stats<<<dim3(1), dim3(128), 0, stream>>>(
      parts, parts + (size_t)nblk * 64, nblk, mtot, g0, be0, stats + 0, stats + 128, 64);

  // layer 1: 64 -> 64 (fuses BN0+ReLU on input)
  gemm_bn<64, 64, 64, true><<<dim3(nblk), dim3(256), 0, stream>>>(
      Y1, w1, b1, Y2, parts, parts + (size_t)nblk * 64, stats + 0, stats + 128);
  finalize_stats<<<dim3(1), dim3(128), 0, stream>>>(
      parts, parts + (size_t)nblk * 64, nblk, mtot, g1, be1, stats + 256, stats + 384, 64);

  // layer 2: 64 -> 128 (fuses BN1+ReLU on input)
  gemm_bn<64, 64, 128, true><<<dim3(nblk), dim3(256), 0, stream>>>(
      Y2, w2, b2, Y3, parts, parts + (size_t)nblk * 128, stats + 256, stats + 384);
  finalize_stats<<<dim3(1), dim3(128), 0, stream>>>(
      parts, parts + (size_t)nblk * 128, nblk, mtot, g2, be2, stats + 512, stats + 640, 128);

  // BN2 + ReLU + max over samples
  pool_kernel<<<dim3((NB * NPOINT) / 8), dim3(256), 0, stream>>>(Y3, stats + 512, stats + 640, out_pool);
}